// DeepFMModel_37014028157491
// MI455X (gfx1250) — compile-verified
//
#include <hip/hip_runtime.h>
#include <hip/hip_bf16.h>
#include <math.h>

// ---------------------------------------------------------------------------
// DeepFM, algebraically reduced:
//   fm_term[b] = x_b^T M x_b   with  M[j1,j2] = dot4(fm_w[j1],fm_w[j2]) * wo[pair(j1,j2)]  (j1<j2)
//   h1 = relu(X @ w1 + b1); h2 = relu(h1 @ w2 + b2)
//   out[b] = sigmoid(fm_term[b] + dot(h2[b], wo[NPAIRS:]) + bo)
// All GEMMs run on V_WMMA_F32_16X16X4_F32 (full f32 matrix core, wave32).
// ---------------------------------------------------------------------------

#define NFEAT   512
#define BATCH   512
#define HDIM    128
#define NPAIRS  (NFEAT * (NFEAT - 1) / 2)   // 130816

typedef __attribute__((ext_vector_type(2))) float v2f;
typedef __attribute__((ext_vector_type(8))) float v8f;

// ---------------------------------------------------------------------------
// Kernel 1: build M (512x512, strictly upper triangular) and zero fm_term.
// ---------------------------------------------------------------------------
__global__ void build_M_kernel(const float* __restrict__ fm_w,
                               const float* __restrict__ wo,
                               float* __restrict__ Mm,
                               float* __restrict__ fm_term) {
    int idx = blockIdx.x * blockDim.x + threadIdx.x;   // 0 .. 512*512-1
    int j1 = idx >> 9;
    int j2 = idx & (NFEAT - 1);
    float v = 0.0f;
    if (j1 < j2) {
        float d = 0.0f;
        #pragma unroll
        for (int k = 0; k < 4; ++k)
            d += fm_w[j1 * 4 + k] * fm_w[j2 * 4 + k];
        // row-major upper-triangle pair index (matches np.triu_indices order)
        int p = j1 * (2 * NFEAT - j1 - 1) / 2 + (j2 - j1 - 1);
        v = d * wo[p];
    }
    Mm[idx] = v;
    if (idx < BATCH) fm_term[idx] = 0.0f;
}

// ---------------------------------------------------------------------------
// Kernel 2: Y = X @ M fused with fm_term[b] += dot(Y[b,:], x[b,:]).
// One wave per 16x16 output tile; K swept 4 at a time via wmma_f32_16x16x4_f32.
// Never materializes Y.
// ---------------------------------------------------------------------------
__global__ void gemm_fm_kernel(const float* __restrict__ X,
                               const float* __restrict__ Mm,
                               float* __restrict__ fm_term) {
    const int lane = threadIdx.x & 31;
    const int wave = threadIdx.x >> 5;
    const int tile = blockIdx.x * (blockDim.x >> 5) + wave;
    const int ntn  = NFEAT / 16;          // 32 column tiles
    const int bt   = tile / ntn;          // batch-row tile
    const int nt   = tile % ntn;          // output-column tile

    const int arow = bt * 16 + (lane & 15);     // A: M index held by this lane
    const int koff = (lane >> 4) * 2;           // A/B: lane-half K offset {0,2}
    const int ncol = nt * 16 + (lane & 15);     // B/C/D: N index held by this lane

    v8f c = {};
    const float* __restrict__ Abase = X + arow * NFEAT + koff;
    #pragma unroll 4
    for (int k0 = 0; k0 < NFEAT; k0 += 4) {
        // A 16x4 f32: lane holds (M=arow, K=k0+koff .. k0+koff+1)
        v2f a = *(const v2f*)(Abase + k0);
        // B 4x16 f32: lane holds (K=k0+koff .. +1, N=ncol)
        v2f b;
        b.x = Mm[(k0 + koff)     * NFEAT + ncol];
        b.y = Mm[(k0 + koff + 1) * NFEAT + ncol];
        c = __builtin_amdgcn_wmma_f32_16x16x4_f32(
                false, a, false, b, (short)0, c, false, false);
    }

    // Fused epilogue: fm_term[b] += sum_n Ytile[m][n] * X[b][n0+n]
    // C/D layout: c[r] on lane L is (M = r + 8*(L>>4), N = L&15) within tile.
    #pragma unroll
    for (int r = 0; r < 8; ++r) {
        int m = bt * 16 + r + ((lane >> 4) << 3);
        float v = c[r] * X[m * NFEAT + ncol];
        // reduce across the 16 lanes sharing this row (xor within low 4 bits)
        #pragma unroll
        for (int s = 1; s < 16; s <<= 1) v += __shfl_xor(v, s, 32);
        if ((lane & 15) == 0) atomicAdd(&fm_term[m], v);
    }
}

// ---------------------------------------------------------------------------
// Kernel 3: C = relu(A @ B + bias), generic row-major f32 WMMA GEMM.
// A: [Md x Kd] lda=Kd, B: [Kd x Nd] ldb=Nd, C: [Md x Nd].
// ---------------------------------------------------------------------------
__global__ void gemm_bias_relu_kernel(const float* __restrict__ A,
                                      const float* __restrict__ B,
                                      const float* __restrict__ bias,
                                      float* __restrict__ C,
                                      int Nd, int Kd) {
    const int lane = threadIdx.x & 31;
    const int wave = threadIdx.x >> 5;
    const int tile = blockIdx.x * (blockDim.x >> 5) + wave;
    const int ntn  = Nd / 16;
    const int bt   = tile / ntn;
    const int nt   = tile % ntn;

    const int arow = bt * 16 + (lane & 15);
    const int koff = (lane >> 4) * 2;
    const int ncol = nt * 16 + (lane & 15);

    v8f c = {};
    const float* __restrict__ Abase = A + arow * Kd + koff;
    #pragma unroll 4
    for (int k0 = 0; k0 < Kd; k0 += 4) {
        v2f a = *(const v2f*)(Abase + k0);
        v2f b;
        b.x = B[(k0 + koff)     * Nd + ncol];
        b.y = B[(k0 + koff + 1) * Nd + ncol];
        c = __builtin_amdgcn_wmma_f32_16x16x4_f32(
                false, a, false, b, (short)0, c, false, false);
    }

    float bv = bias[ncol];
    #pragma unroll
    for (int r = 0; r < 8; ++r) {
        int m = bt * 16 + r + ((lane >> 4) << 3);
        float v = c[r] + bv;
        C[m * Nd + ncol] = v > 0.0f ? v : 0.0f;
    }
}

// ---------------------------------------------------------------------------
// Kernel 4: out[b] = sigmoid(fm_term[b] + dot(H2[b,:], wo[NPAIRS:]) + bo).
// One wave per batch row.
// ---------------------------------------------------------------------------
__global__ void final_kernel(const float* __restrict__ fm_term,
                             const float* __restrict__ H2,
                             const float* __restrict__ wo,
                             const float* __restrict__ bo,
                             float* __restrict__ out) {
    const int lane = threadIdx.x & 31;
    const int wave = threadIdx.x >> 5;
    const int b    = blockIdx.x * (blockDim.x >> 5) + wave;
    const float* __restrict__ wod = wo + NPAIRS;

    float v = 0.0f;
    #pragma unroll
    for (int h = lane; h < HDIM; h += 32)
        v += H2[b * HDIM + h] * wod[h];
    #pragma unroll
    for (int s = 1; s < 32; s <<= 1) v += __shfl_xor(v, s, 32);

    if (lane == 0) {
        float z = fm_term[b] + v + bo[0];
        out[b] = 1.0f / (1.0f + expf(-z));
    }
}

// ---------------------------------------------------------------------------
// Launch
// ---------------------------------------------------------------------------
extern "C" void kernel_launch(void* const* d_in, const int* in_sizes, int n_in,
                              void* d_out, int out_size, void* d_ws, size_t ws_size,
                              hipStream_t stream) {
    (void)in_sizes; (void)n_in; (void)out_size; (void)ws_size;

    const float* x    = (const float*)d_in[0];  // [512, 512]
    const float* fm_w = (const float*)d_in[1];  // [512, 4]
    const float* w1   = (const float*)d_in[2];  // [512, 128]
    const float* b1   = (const float*)d_in[3];  // [128]
    const float* w2   = (const float*)d_in[4];  // [128, 128]
    const float* b2   = (const float*)d_in[5];  // [128]
    const float* wo   = (const float*)d_in[6];  // [130944, 1]
    const float* bo   = (const float*)d_in[7];  // [1]
    float* out = (float*)d_out;                 // [512, 1]

    // Workspace layout (bytes): M 1MB | fm_term 2KB | H1 256KB | H2 256KB
    char* ws = (char*)d_ws;
    float* Mm      = (float*)(ws);
    float* fm_term = (float*)(ws + (size_t)NFEAT * NFEAT * 4);
    float* H1      = (float*)(ws + (size_t)NFEAT * NFEAT * 4 + 2048);
    float* H2      = (float*)(ws + (size_t)NFEAT * NFEAT * 4 + 2048 + (size_t)BATCH * HDIM * 4);

    // 1) Build M, zero fm_term accumulators.
    build_M_kernel<<<(NFEAT * NFEAT) / 256, 256, 0, stream>>>(fm_w, wo, Mm, fm_term);

    // 2) FM quadratic form: (512x512 tiles) = 1024 tiles, 8 waves/block -> 128 blocks.
    gemm_fm_kernel<<<128, 256, 0, stream>>>(x, Mm, fm_term);

    // 3) H1 = relu(X @ w1 + b1): 32x8 = 256 tiles -> 32 blocks.
    gemm_bias_relu_kernel<<<32, 256, 0, stream>>>(x, w1, b1, H1, HDIM, NFEAT);

    // 4) H2 = relu(H1 @ w2 + b2): 256 tiles -> 32 blocks.
    gemm_bias_relu_kernel<<<32, 256, 0, stream>>>(H1, w2, b2, H2, HDIM, HDIM);

    // 5) Final: 512 rows, 8 waves/block -> 64 blocks.
    final_kernel<<<64, 256, 0, stream>>>(fm_term, H2, wo, bo, out);
}